// ResamplePointMip_34076270527095
// MI455X (gfx1250) — compile-verified
//
#include <hip/hip_runtime.h>
#include <stdint.h>

// ---------------------------------------------------------------------------
// ResamplePointMip for MI455X (gfx1250), memory-bound kernel.
//   inputs : rays (N,6) f32 | radii (N,1) f32 | weights (N,64,1) f32 | t_vals (N,65) f32
//   outputs: new_t (N,65) | means (N,64,3) | covs (N,64,3)   (flat-concatenated)
// Strategy: 1 ray per thread; ray-contiguous weights/t_vals are staged with
// CDNA5 async global->LDS loads into a TRANSPOSED [elem][ray] LDS layout so
// compute-phase LDS reads are bank-conflict-free and staging is coalesced.
// Inverse-CDF sampling uses a two-pointer merge (u and cdf both sorted).
// ---------------------------------------------------------------------------

#define TPB   64                       // threads (= rays) per block, 2 wave32s
#define NBINS 64                       // weights per ray
#define NT    65                       // t_vals / samples per ray
#define F32_EPS 1.1920928955078125e-07f
#define RESAMPLE_PAD 0.01f

// Low 32 bits of a generic pointer to __shared__ == LDS byte address.
__device__ __forceinline__ uint32_t lds_lo32(const void* p) {
  return (uint32_t)(uintptr_t)p;
}

// CDNA5 async DMA: global -> LDS, one dword per lane, tracked by ASYNCcnt.
__device__ __forceinline__ void async_g2l_b32(uint32_t lds_off, const float* g) {
  asm volatile("global_load_async_to_lds_b32 %0, %1, off"
               :: "v"(lds_off), "v"((uint64_t)(uintptr_t)g)
               : "memory");
}

__device__ __forceinline__ void wait_asynccnt0() {
  asm volatile("s_wait_asynccnt 0x0" ::: "memory");
}

__global__ __launch_bounds__(TPB) void resample_point_mip_kernel(
    const float* __restrict__ rays,     // (N,6)
    const float* __restrict__ radii,    // (N,1)
    const float* __restrict__ weights,  // (N,64)
    const float* __restrict__ tvals,    // (N,65)
    float* __restrict__ out_t,          // (N,65)
    float* __restrict__ out_mean,       // (N,64,3)
    float* __restrict__ out_cov,        // (N,64,3)
    int N)
{
  // Column-major LDS: element i of local ray r lives at [i*TPB + r].
  // Lanes access consecutive addresses -> conflict-free across 64 banks.
  __shared__ float sW[NBINS * TPB];     // weights, then blurred weights
  __shared__ float sT[NT * TPB];        // bins (t_vals)
  __shared__ float sC[NT * TPB];        // cdf

  const int tid  = threadIdx.x;
  const int ray0 = blockIdx.x * TPB;
  const int nray = min(TPB, N - ray0);
  const int gray = ray0 + tid;

  if (gray < N) {                       // overlap: pull ray geometry toward L2/L0
    __builtin_prefetch(rays + (size_t)gray * 6, 0, 3);   // global_prefetch_b8
    __builtin_prefetch(radii + gray, 0, 3);
  }

  // ---- async-stage weights, transposing (coalesced global, scattered LDS) --
  {
    const float* gw = weights + (size_t)ray0 * NBINS;
    const int tot = nray * NBINS;
    for (int e = tid; e < tot; e += TPB) {
      int r = e >> 6;                   // local ray
      int i = e & (NBINS - 1);          // bin
      async_g2l_b32(lds_lo32(&sW[i * TPB + r]), gw + e);
    }
  }
  // ---- async-stage t_vals, transposing ------------------------------------
  {
    const float* gt = tvals + (size_t)ray0 * NT;
    const int tot = nray * NT;
    for (int e = tid; e < tot; e += TPB) {
      int r = (int)((unsigned)e / NT);
      int i = e - r * NT;
      async_g2l_b32(lds_lo32(&sT[i * TPB + r]), gt + e);
    }
  }
  wait_asynccnt0();                     // this wave's async DMAs complete
  __syncthreads();                      // cross-wave visibility of LDS

  if (tid < nray) {
    const float* rp = rays + (size_t)gray * 6;
    const float ox = rp[0], oy = rp[1], oz = rp[2];
    const float dx = rp[3], dy = rp[4], dz = rp[5];
    const float rad = radii[gray];

    // ---- max-blur of weights + padding (rolling window, in-place) ---------
    float wim1 = sW[tid];               // w[-1] clamps to w[0]
    float wi   = wim1;
    float wsum = 0.0f;
#pragma unroll 4
    for (int i = 0; i < NBINS; ++i) {
      float wip1 = (i < NBINS - 1) ? sW[(i + 1) * TPB + tid] : wi;  // w[64]->w[63]
      float wb = 0.5f * (fmaxf(wim1, wi) + fmaxf(wi, wip1)) + RESAMPLE_PAD;
      sW[i * TPB + tid] = wb;           // safe: originals carried in registers
      wsum += wb;
      wim1 = wi; wi = wip1;
    }

    const float pad  = fmaxf(0.0f, 1e-5f - wsum);
    const float wadd = pad * (1.0f / NBINS);
    const float inv  = 1.0f / (wsum + pad);

    // ---- cdf: [0, clamped cumsum(pdf[0:63]), 1] ---------------------------
    float csum = 0.0f;
    sC[tid] = 0.0f;
#pragma unroll 4
    for (int i = 1; i < NT - 1; ++i) {
      csum += (sW[(i - 1) * TPB + tid] + wadd) * inv;   // pdf[i-1]
      sC[i * TPB + tid] = fminf(1.0f, csum);
    }
    sC[(NT - 1) * TPB + tid] = 1.0f;

    // ---- ray geometry shared across all 64 intervals ----------------------
    const float rvar = 0.25f * rad * rad;
    const float dmag = fmaxf(1e-10f, dx * dx + dy * dy + dz * dz);
    const float d2x = dx * dx, d2y = dy * dy, d2z = dz * dz;
    const float xcx = rvar * (1.0f - d2x / dmag);
    const float xcy = rvar * (1.0f - d2y / dmag);
    const float xcz = rvar * (1.0f - d2z / dmag);

    float* ot = out_t   + (size_t)gray * NT;
    float* om = out_mean + (size_t)gray * (NBINS * 3);
    float* oc = out_cov  + (size_t)gray * (NBINS * 3);

    // ---- inverse-CDF resample: two-pointer merge over sorted u & cdf ------
    const float ustep = (1.0f - F32_EPS) / 64.0f;       // linspace(0, 1-eps, 65)
    int   i0 = 0;
    float tprev = 0.0f;
    for (int s = 0; s < NT; ++s) {
      const float u = (float)s * ustep;
      while (i0 < NBINS - 1 && sC[(i0 + 1) * TPB + tid] <= u) ++i0;
      const float c0 = sC[i0 * TPB + tid];
      const float c1 = sC[(i0 + 1) * TPB + tid];        // > u >= c0 by construction
      const float den = c1 - c0;
      // reference: nan_to_num((u-c0)/den, 0) then clip(0,1); den>0 always here
      const float t = (den > 0.0f)
                        ? fminf(1.0f, fmaxf(0.0f, (u - c0) / den))
                        : 0.0f;
      const float b0 = sT[i0 * TPB + tid];
      const float b1 = sT[(i0 + 1) * TPB + tid];
      const float tv = b0 + t * (b1 - b0);
      ot[s] = tv;
      if (s > 0) {                                       // interval k = s-1
        const float tm   = 0.5f * (tprev + tv);
        const float dt   = tv - tprev;
        const float tvar = dt * dt * (1.0f / 12.0f);
        const int k = (s - 1) * 3;
        om[k + 0] = dx * tm + ox;
        om[k + 1] = dy * tm + oy;
        om[k + 2] = dz * tm + oz;
        oc[k + 0] = tvar * d2x + xcx;
        oc[k + 1] = tvar * d2y + xcy;
        oc[k + 2] = tvar * d2z + xcz;
      }
      tprev = tv;
    }
  }
}

extern "C" void kernel_launch(void* const* d_in, const int* in_sizes, int n_in,
                              void* d_out, int out_size, void* d_ws, size_t ws_size,
                              hipStream_t stream) {
  const float* rays    = (const float*)d_in[0];
  const float* radii   = (const float*)d_in[1];
  const float* weights = (const float*)d_in[2];   // (N,64,1) flat
  const float* tvals   = (const float*)d_in[3];
  const int N = in_sizes[0] / 6;

  float* out   = (float*)d_out;
  float* out_t = out;                              // N*65
  float* out_m = out_t + (size_t)N * NT;           // N*192
  float* out_c = out_m + (size_t)N * NBINS * 3;    // N*192

  const int grid = (N + TPB - 1) / TPB;
  hipLaunchKernelGGL(resample_point_mip_kernel, dim3(grid), dim3(TPB), 0, stream,
                     rays, radii, weights, tvals, out_t, out_m, out_c, N);
}